// CustomCNNLoss_54159537602783
// MI455X (gfx1250) — compile-verified
//
#include <hip/hip_runtime.h>
#include <hip/hip_bf16.h>
#include <math.h>

// ---- problem constants (from reference) ----
#define S_GRID 28
#define BB 2
#define NC 80
#define CELLS_PER_IMG (S_GRID * S_GRID)   // 784
#define BATCH 512
#define NCELLS (BATCH * CELLS_PER_IMG)    // 401408
#define NTGT 50
#define CH (BB * 5 + NC)                  // 90
#define LAMBDA_COORD 5.0f
#define LAMBDA_NOOBJ 0.1f
#define EPS_IOU 1e-6f
#define EPSF 1e-7f
#define PI_F 3.14159265358979323846f

typedef __attribute__((ext_vector_type(2))) float v2f;
typedef __attribute__((ext_vector_type(8))) float v8f;

// ---------------- kernel 0: zero workspace state ----------------
__global__ void zero_ws_kernel(float* __restrict__ p, int n) {
    int i = blockIdx.x * blockDim.x + threadIdx.x;
    if (i < n) p[i] = 0.0f;
}

// ---------------- kernel 1: sequential target assignment ----------------
// One thread per batch image replays the jax.lax.scan exactly (order matters:
// strict-greater IoU replacement, first-max argmax tie break).
__global__ void build_targets_kernel(const float* __restrict__ pred,
                                     const float* __restrict__ tgt,
                                     float* __restrict__ gtb,
                                     float* __restrict__ biou,
                                     int* __restrict__ flags,
                                     unsigned* __restrict__ clsm) {
    int b = blockIdx.x * blockDim.x + threadIdx.x;
    if (b >= BATCH) return;
    const float cell = 1.0f / (float)S_GRID;
    for (int t = 0; t < NTGT; ++t) {
        const float* g = tgt + ((size_t)b * NTGT + t) * 5;
        float cls = g[0];
        if (cls < 0.0f) continue;                 // invalid slot
        int cid = (int)cls;
        cid = cid < 0 ? 0 : (cid > NC - 1 ? NC - 1 : cid);
        float cx = g[1], cy = g[2], w = g[3], h = g[4];
        int col = (int)(cx / cell); if (col > S_GRID - 1) col = S_GRID - 1;
        int row = (int)(cy / cell); if (row > S_GRID - 1) row = S_GRID - 1;
        float cxr = cx / cell - (float)col;
        float cyr = cy / cell - (float)row;
        int idx = b * CELLS_PER_IMG + row * S_GRID + col;
        const float* pr = pred + (size_t)idx * CH;

        float gx1 = cx - w * 0.5f, gy1 = cy - h * 0.5f;
        float gx2 = cx + w * 0.5f, gy2 = cy + h * 0.5f;
        float garea = fmaxf(gx2 - gx1, 0.f) * fmaxf(gy2 - gy1, 0.f);

        float best = -1.0f; int bj = 0;
        #pragma unroll
        for (int j = 0; j < BB; ++j) {
            float px = (pr[j * 5 + 0] + (float)col) * cell;
            float py = (pr[j * 5 + 1] + (float)row) * cell;
            float pw = pr[j * 5 + 2], ph = pr[j * 5 + 3];
            float px1 = px - pw * 0.5f, py1 = py - ph * 0.5f;
            float px2 = px + pw * 0.5f, py2 = py + ph * 0.5f;
            float iw = fmaxf(fminf(px2, gx2) - fmaxf(px1, gx1), 0.f);
            float ih = fmaxf(fminf(py2, gy2) - fmaxf(py1, gy1), 0.f);
            float inter = iw * ih;
            float uni = fmaxf(px2 - px1, 0.f) * fmaxf(py2 - py1, 0.f) + garea - inter;
            float iou = inter / (uni + EPS_IOU);
            if (iou > best) { best = iou; bj = j; }  // first-max tie break
        }
        int fl = flags[idx];
        bool occ = (fl & 1) != 0;
        bool upd = (!occ) || (best > biou[idx]);     // strict >
        if (upd) {
            gtb[(size_t)idx * 4 + 0] = cxr;
            gtb[(size_t)idx * 4 + 1] = cyr;
            gtb[(size_t)idx * 4 + 2] = w;
            gtb[(size_t)idx * 4 + 3] = h;
            biou[idx] = best;
            flags[idx] = 1 | (bj << 1);
        }
        clsm[(size_t)idx * 3 + (cid >> 5)] |= 1u << (cid & 31);  // gt_cls .max(1)
    }
}

// ---------------- kernel 2: per-cell loss + WMMA wave reduction ----------------
__global__ void __launch_bounds__(256) yolo_loss_main_kernel(
        const float* __restrict__ pred,
        const float* __restrict__ gtb,
        const float* __restrict__ biou,
        const int* __restrict__ flags,
        const unsigned* __restrict__ clsm,
        float* __restrict__ partials) {
    const float cell = 1.0f / (float)S_GRID;
    int idx = blockIdx.x * blockDim.x + threadIdx.x;   // grid is exact: no guard
    int b   = idx / CELLS_PER_IMG;
    int rem = idx - b * CELLS_PER_IMG;
    int row = rem / S_GRID;
    int col = rem - row * S_GRID;
    const float* p = pred + (size_t)idx * CH;          // 8-byte aligned (360B stride)

    // 10 box floats as 5x float2
    float pb[10];
    const float2* p2 = (const float2*)p;
    #pragma unroll
    for (int i = 0; i < 5; ++i) { float2 v = p2[i]; pb[2 * i] = v.x; pb[2 * i + 1] = v.y; }

    int fl  = flags[idx];
    int obj = fl & 1;
    int j   = (fl >> 1) & 1;

    float lsum = 0.0f;
    float c0 = pb[4], c1 = pb[9];
    float cr = j ? c1 : c0;

    // no-object confidence term: (~obj)[...,None] | (onehot==0)
    float noobj = obj ? (j ? c0 * c0 : c1 * c1) : (c0 * c0 + c1 * c1);
    lsum += LAMBDA_NOOBJ * noobj;

    if (obj) {
        float bi = biou[idx];
        float d = cr - bi;
        lsum += d * d;                                 // objectness term

        // ---- CIoU ----
        float gx = (gtb[(size_t)idx * 4 + 0] + (float)col) * cell;
        float gy = (gtb[(size_t)idx * 4 + 1] + (float)row) * cell;
        float gw = gtb[(size_t)idx * 4 + 2], gh = gtb[(size_t)idx * 4 + 3];
        float px = (pb[j * 5 + 0] + (float)col) * cell;
        float py = (pb[j * 5 + 1] + (float)row) * cell;
        float pw = fabsf(pb[j * 5 + 2]), ph = fabsf(pb[j * 5 + 3]);
        float px1 = px - pw * 0.5f, py1 = py - ph * 0.5f, px2 = px + pw * 0.5f, py2 = py + ph * 0.5f;
        float gx1 = gx - gw * 0.5f, gy1 = gy - gh * 0.5f, gx2 = gx + gw * 0.5f, gy2 = gy + gh * 0.5f;
        float iw = fmaxf(fminf(px2, gx2) - fmaxf(px1, gx1), 0.f);
        float ih = fmaxf(fminf(py2, gy2) - fmaxf(py1, gy1), 0.f);
        float inter = iw * ih;
        float uni = fmaxf(px2 - px1, 0.f) * fmaxf(py2 - py1, 0.f)
                  + fmaxf(gx2 - gx1, 0.f) * fmaxf(gy2 - gy1, 0.f) - inter;
        float iou = inter / (uni + EPSF);
        float dx = px - gx, dy = py - gy;
        float rho2 = dx * dx + dy * dy;
        float cw  = fmaxf(px2, gx2) - fminf(px1, gx1);
        float chh = fmaxf(py2, gy2) - fminf(py1, gy1);
        float c2  = cw * cw + chh * chh + EPSF;
        float da  = atanf(gw / (gh + EPSF)) - atanf(pw / (ph + EPSF));
        float v   = (4.0f / (PI_F * PI_F)) * da * da;
        float alpha = v / (1.0f - iou + v + EPSF);     // stop_gradient: value only
        lsum += LAMBDA_COORD * (1.0f - iou + rho2 / c2 + alpha * v);

        // ---- class BCE (only obj cells touch the 80-channel tail) ----
        float bce = 0.0f;
        #pragma unroll
        for (int wI = 0; wI < 3; ++wI) {
            unsigned m = clsm[(size_t)idx * 3 + wI];
            int n = (wI == 2) ? 16 : 32;
            for (int k = 0; k < n; ++k) {
                float pv = p[BB * 5 + wI * 32 + k];
                float gv = (float)((m >> k) & 1u);
                bce += fmaxf(pv, 0.f) - pv * gv + log1pf(__expf(-fabsf(pv)));
            }
        }
        lsum += bce;
    }

    // ---- wave32 reduction on the matrix pipe (EXEC all ones here) ----
    // A(16x4): lanes 0-15 -> A[m,0]=lsum, lanes 16-31 -> A[m,2]=lsum, K=1,3 zero.
    // B = ones => D[m,n] = lsum[m] + lsum[m+16] (layout-proof w.r.t. B).
    v2f a;  a.x = lsum; a.y = 0.0f;
    v2f bo; bo.x = 1.0f; bo.y = 1.0f;
    v8f acc = {};
    acc = __builtin_amdgcn_wmma_f32_16x16x4_f32(
        /*neg_a=*/false, a, /*neg_b=*/false, bo,
        /*c_mod=*/(short)0, acc, /*reuse_a=*/false, /*reuse_b=*/false);
    float s8 = acc[0] + acc[1] + acc[2] + acc[3] + acc[4] + acc[5] + acc[6] + acc[7];
    // lanes 0-15 hold sum over M=0..7, lanes 16-31 hold sum over M=8..15
    float wave_sum = s8 + __shfl_xor(s8, 16, 32);

    __shared__ float wpart[8];
    int lane = threadIdx.x & 31, wave = threadIdx.x >> 5;
    if (lane == 0) wpart[wave] = wave_sum;
    __syncthreads();
    if (threadIdx.x == 0) {
        float t = 0.f;
        #pragma unroll
        for (int i = 0; i < 8; ++i) t += wpart[i];
        partials[blockIdx.x] = t;
    }
}

// ---------------- kernel 3: deterministic final reduction ----------------
__global__ void final_reduce_kernel(const float* __restrict__ partials,
                                    float* __restrict__ out, int n) {
    __shared__ float sm[256];
    float t = 0.f;
    for (int i = threadIdx.x; i < n; i += 256) t += partials[i];
    sm[threadIdx.x] = t;
    __syncthreads();
    for (int s = 128; s > 0; s >>= 1) {
        if (threadIdx.x < s) sm[threadIdx.x] += sm[threadIdx.x + s];
        __syncthreads();
    }
    if (threadIdx.x == 0) out[0] = sm[0] / (float)BATCH;
}

// ---------------- launcher ----------------
extern "C" void kernel_launch(void* const* d_in, const int* in_sizes, int n_in,
                              void* d_out, int out_size, void* d_ws, size_t ws_size,
                              hipStream_t stream) {
    const float* pred = (const float*)d_in[0];   // (512,28,28,90) f32
    const float* tgt  = (const float*)d_in[1];   // (512,50,5)     f32

    // workspace layout (SoA), ~14.5 MB total
    float*    gtb      = (float*)d_ws;                         // 4*NCELLS
    float*    biou     = gtb + 4 * (size_t)NCELLS;             // NCELLS
    int*      flags    = (int*)(biou + NCELLS);                // NCELLS
    unsigned* clsm     = (unsigned*)(flags + NCELLS);          // 3*NCELLS
    float*    partials = (float*)(clsm + 3 * (size_t)NCELLS);  // NCELLS/256

    int zn = 9 * NCELLS;   // dwords of state to clear (partials fully overwritten)
    zero_ws_kernel<<<(zn + 255) / 256, 256, 0, stream>>>((float*)d_ws, zn);

    build_targets_kernel<<<(BATCH + 63) / 64, 64, 0, stream>>>(
        pred, tgt, gtb, biou, flags, clsm);

    int nblocks = NCELLS / 256;   // 1568, exact
    yolo_loss_main_kernel<<<nblocks, 256, 0, stream>>>(
        pred, gtb, biou, flags, clsm, partials);

    final_reduce_kernel<<<1, 256, 0, stream>>>(partials, (float*)d_out, nblocks);
}